// CRFParallel_89747636617708
// MI455X (gfx1250) — compile-verified
//
#include <hip/hip_runtime.h>
#include <hip/hip_bf16.h>

// CRF negative log-likelihood, CDNA5 (gfx1250).
// Forward recurrence done as exp-domain f32 WMMA matmul per time step:
//   new_p[b,j] = rowmax[b] + log( sum_i exp(p[b,i]-rowmax[b]) * E[i,j] ) + feats[t,b,j]
// with E = exp(transitions) (constant; -10000 entries underflow to exact 0).

#define TT 512
#define BB 64
#define LL 48
#define PSTR 49            // LDS row stride (coprime with 64 banks)
#define START_TAG 46
#define STOP_TAG  47
#define NTHREADS 384       // 12 wave32s -> 4x3 grid of 16x16 output tiles

typedef float v2f __attribute__((ext_vector_type(2)));
typedef float v8f __attribute__((ext_vector_type(8)));

__global__ __launch_bounds__(NTHREADS)
void crf_nll_kernel(const float* __restrict__ feats,        // (T,B,L)
                    const float* __restrict__ trans,        // (L,L)
                    const int* __restrict__ tags,           // (B,T)
                    const unsigned char* __restrict__ mask, // (B,T) bool
                    float* __restrict__ out)                // scalar
{
    __shared__ float sT[LL * PSTR];   // raw transitions
    __shared__ float sE[LL * PSTR];   // exp(transitions)
    __shared__ float sP[BB * PSTR];   // partition (log domain)
    __shared__ float sA[BB * PSTR];   // exp(p - rowmax)
    __shared__ float sMax[BB];
    __shared__ float sRed[NTHREADS];

    const int tid  = threadIdx.x;
    const int lane = tid & 31;
    const int wave = tid >> 5;        // 0..11
    const int bt   = wave / 3;        // batch-tile 0..3  (rows 16*bt .. +15)
    const int ct   = wave % 3;        // label-tile 0..2  (cols 16*ct .. +15)
    const int l15  = lane & 15;
    const int k0   = (lane < 16) ? 0 : 2;   // K pair held by this half-wave
    const int roff = (lane < 16) ? 0 : 8;   // C/D row offset for high lanes

    // ---- load transitions, build E = exp(T) ----
    for (int idx = tid; idx < LL * LL; idx += NTHREADS) {
        int i = idx / LL, j = idx % LL;
        float v = trans[idx];
        sT[i * PSTR + j] = v;
        sE[i * PSTR + j] = __expf(v);
    }
    __syncthreads();

    // ---- init partition: p = feats[0] + T[START,:] ----
    for (int idx = tid; idx < BB * LL; idx += NTHREADS) {
        int b = idx / LL, j = idx % LL;
        sP[b * PSTR + j] = feats[idx] + sT[START_TAG * PSTR + j];
    }

    // ---- preload loop-invariant B fragments of E for this wave's column tile ----
    // B 4x16 layout: lanes 0-15 hold K=0,1 (VGPR0,1); lanes 16-31 hold K=2,3.
    v2f e_frag[12];
    {
        const int n = ct * 16 + l15;
        #pragma unroll
        for (int kk = 0; kk < 12; ++kk) {
            e_frag[kk].x = sE[(4 * kk + k0) * PSTR + n];
            e_frag[kk].y = sE[(4 * kk + k0 + 1) * PSTR + n];
        }
    }
    __syncthreads();

    // ---- sequential forward scan over t = 1..T-1 ----
    for (int t = 1; t < TT; ++t) {
        const float* __restrict__ ft = feats + (size_t)t * BB * LL;

        // prefetch next step's emission tile (12 KB) into cache
        if (t + 1 < TT && tid < 96) {
            __builtin_prefetch((const char*)(ft + BB * LL) + tid * 128, 0, 1);
        }

        // phase 1: per-batch row max of p
        if (tid < BB) {
            float m = -__builtin_inff();
            #pragma unroll 8
            for (int i = 0; i < LL; ++i) m = fmaxf(m, sP[tid * PSTR + i]);
            sMax[tid] = m;
        }
        __syncthreads();

        // phase 2: A = exp(p - rowmax)
        for (int idx = tid; idx < BB * LL; idx += NTHREADS) {
            int b = idx / LL, i = idx % LL;
            sA[b * PSTR + i] = __expf(sP[b * PSTR + i] - sMax[b]);
        }
        __syncthreads();

        // phase 3: C(16x16) = A_tile(16x48) * E_tile(48x16) via 12x WMMA f32 16x16x4
        v8f c = {0.f, 0.f, 0.f, 0.f, 0.f, 0.f, 0.f, 0.f};
        {
            const int m_row = bt * 16 + l15;    // A 16x4: lanes share rows M=0..15
            #pragma unroll
            for (int kk = 0; kk < 12; ++kk) {
                v2f a;
                a.x = sA[m_row * PSTR + 4 * kk + k0];
                a.y = sA[m_row * PSTR + 4 * kk + k0 + 1];
                c = __builtin_amdgcn_wmma_f32_16x16x4_f32(
                        /*neg_a=*/false, a, /*neg_b=*/false, e_frag[kk],
                        /*c_mod=*/(short)0, c, /*reuse_a=*/false, /*reuse_b=*/false);
            }
        }

        // phase 4: p[b,j] = mask ? rowmax[b] + log(c) + feats[t,b,j] : p[b,j]
        {
            const int j = ct * 16 + l15;
            #pragma unroll
            for (int r = 0; r < 8; ++r) {
                const int b = bt * 16 + roff + r;
                const float oldv = sP[b * PSTR + j];
                const float nv = sMax[b] + __logf(c[r]) + ft[b * LL + j];
                sP[b * PSTR + j] = mask[b * TT + t] ? nv : oldv;
            }
        }
        __syncthreads();
    }

    // ---- tail: forward score + gold score, accumulated per thread ----
    float part = 0.f;

    // forward_score = sum_b logsumexp_i( p[b,i] + T[i,STOP] )
    if (tid < BB) {
        float m = -__builtin_inff();
        #pragma unroll 8
        for (int i = 0; i < LL; ++i)
            m = fmaxf(m, sP[tid * PSTR + i] + sT[i * PSTR + STOP_TAG]);
        float s = 0.f;
        #pragma unroll 8
        for (int i = 0; i < LL; ++i)
            s += __expf(sP[tid * PSTR + i] + sT[i * PSTR + STOP_TAG] - m);
        part += m + __logf(s);
    }

    // gold path: emissions + transitions where mask is set
    for (int idx = tid; idx < BB * TT; idx += NTHREADS) {
        if (mask[idx]) {
            const int b = idx / TT, t = idx % TT;
            const int tag  = tags[idx];
            const int prev = (t == 0) ? START_TAG : tags[idx - 1];
            part -= feats[((size_t)t * BB + b) * LL + tag] + sT[prev * PSTR + tag];
        }
    }

    // end energy: T[tags[b, len_b-1], STOP]
    if (tid < BB) {
        int len = 0;
        for (int t = 0; t < TT; ++t) len += mask[tid * TT + t] ? 1 : 0;
        const int end_id = tags[tid * TT + (len - 1)];
        part -= sT[end_id * PSTR + STOP_TAG];
    }

    // ---- reduce 384 partials, write scalar ----
    sRed[tid] = part;
    __syncthreads();
    if (tid < 128) sRed[tid] += sRed[tid + 256];
    __syncthreads();
    for (int s = 128; s > 0; s >>= 1) {
        if (tid < s && tid + s < 256) sRed[tid] += sRed[tid + s];
        __syncthreads();
    }
    if (tid == 0) out[0] = sRed[0];
}

extern "C" void kernel_launch(void* const* d_in, const int* in_sizes, int n_in,
                              void* d_out, int out_size, void* d_ws, size_t ws_size,
                              hipStream_t stream) {
    (void)in_sizes; (void)n_in; (void)out_size; (void)d_ws; (void)ws_size;
    const float*         feats = (const float*)d_in[0];
    const float*         trans = (const float*)d_in[1];
    const int*           tags  = (const int*)d_in[2];
    const unsigned char* mask  = (const unsigned char*)d_in[3];
    float*               out   = (float*)d_out;
    crf_nll_kernel<<<dim3(1), dim3(NTHREADS), 0, stream>>>(feats, trans, tags, mask, out);
}